// GumbelSoftmaxDiscretization_9088150798858
// MI455X (gfx1250) — compile-verified
//
#include <hip/hip_runtime.h>
#include <hip/hip_bf16.h>
#include <math.h>

#define KCODES 256
#define BLOCK_THREADS 256
#define WAVES_PER_BLOCK 8
#define ROWS_PER_BLOCK 512
#define ROWS_PER_WAVE (ROWS_PER_BLOCK / WAVES_PER_BLOCK) /* 64 */
#define PIPE 4 /* async LDS buffers per wave; 3 rows (3KB) in flight */

// ---- CDNA5 async global->LDS helpers (gfx1250) -----------------------------
// LDS byte offset of a generic pointer to __shared__: low 32 bits of the
// generic (flat) address are the workgroup-relative LDS offset on AMDGPU.
__device__ __forceinline__ unsigned lds_off_u32(const void* p) {
    return (unsigned)(unsigned long long)p;
}

// Issue one 1KB Gumbel row: each of the 32 lanes copies 2 x 16B.
// GVS mode: mem addr = SGPR64 base + VGPR32 offset.  LDS dest = per-lane VGPR address.
__device__ __forceinline__ void issue_gumbel_row(unsigned long long gbase,
                                                 unsigned mem_off,
                                                 unsigned lds_base) {
    asm volatile(
        "global_load_async_to_lds_b128 %0, %2, %4\n\t"
        "global_load_async_to_lds_b128 %1, %3, %4"
        :
        : "v"(lds_base), "v"(lds_base + 512u),
          "v"(mem_off), "v"(mem_off + 512u),
          "s"(gbase)
        : "memory");
}

__device__ __forceinline__ void wait_async_le0() {
    asm volatile("s_wait_asynccnt 0x0" ::: "memory");
}
__device__ __forceinline__ void wait_async_le2() {
    asm volatile("s_wait_asynccnt 0x2" ::: "memory");
}
__device__ __forceinline__ void wait_async_le4() {
    asm volatile("s_wait_asynccnt 0x4" ::: "memory");
}
__device__ __forceinline__ void wait_async_le6() {
    asm volatile("s_wait_asynccnt 0x6" ::: "memory");
}

// ---- main kernel -----------------------------------------------------------
__global__ void __launch_bounds__(BLOCK_THREADS)
gumbel_vq_main(const float* __restrict__ z,
               const float* __restrict__ codebook,
               const float* __restrict__ log_temp,
               const float* __restrict__ gumbel,
               float* __restrict__ out,      // [0,N) disc | [N] perp | [N+1,2N+1) idx
               float* __restrict__ partial,  // [gridDim.x * 256]
               int N) {
    // 8 waves x 4 buffers x 1KB = 32KB; reused after the loop for the
    // per-block avg-probs merge (needs only 8KB).
    __shared__ __align__(16) float smem[WAVES_PER_BLOCK * PIPE * KCODES];
    float (*gbuf)[PIPE][KCODES] = (float (*)[PIPE][KCODES])smem;
    float (*accbuf)[KCODES]     = (float (*)[KCODES])smem;

    const int tid  = threadIdx.x;
    const int w    = tid >> 5;
    const int lane = tid & 31;

    const float inv_tau = __expf(-log_temp[0]);  // 1/exp(log_t)

    // This lane's 8 codebook entries: k = 4*lane+j (j<4) and 128+4*lane+(j-4)
    const float4 c0 = *(const float4*)(codebook + 4 * lane);
    const float4 c1 = *(const float4*)(codebook + 128 + 4 * lane);
    float cb[8] = {c0.x, c0.y, c0.z, c0.w, c1.x, c1.y, c1.z, c1.w};
    int   kk[8];
#pragma unroll
    for (int j = 0; j < 8; ++j)
        kk[j] = (j < 4) ? (4 * lane + j) : (128 + 4 * lane + (j - 4));

    float acc[8] = {0.f, 0.f, 0.f, 0.f, 0.f, 0.f, 0.f, 0.f};

    unsigned ldsb[PIPE];
#pragma unroll
    for (int b = 0; b < PIPE; ++b)
        ldsb[b] = lds_off_u32(&gbuf[w][b][0]) + (unsigned)(lane * 16);

    const int row0 = blockIdx.x * ROWS_PER_BLOCK + w * ROWS_PER_WAVE;
    const unsigned long long gbase = (unsigned long long)gumbel;
    const unsigned lane_byte = (unsigned)(lane * 16);

    // coalesced z preload: this wave's 64 rows, broadcast per-row via shuffle
    const float zA = z[row0 + lane];
    const float zB = z[row0 + 32 + lane];

    // prologue: rows 0..2 -> buffers 0..2 (3KB in flight)
#pragma unroll
    for (int b = 0; b < PIPE - 1; ++b)
        issue_gumbel_row(gbase, (unsigned)(row0 + b) * 1024u + lane_byte, ldsb[b]);

    for (int i = 0; i < ROWS_PER_WAVE; ++i) {
        const int row = row0 + i;
        if (i + (PIPE - 1) < ROWS_PER_WAVE) {
            issue_gumbel_row(gbase,
                             (unsigned)(row + (PIPE - 1)) * 1024u + lane_byte,
                             ldsb[(i + (PIPE - 1)) & (PIPE - 1)]);
            wait_async_le6();                         // row i done, 3 in flight
        } else {
            const int remaining = ROWS_PER_WAVE - 1 - i;  // rows still in flight after i
            if (remaining == 2)      wait_async_le4();
            else if (remaining == 1) wait_async_le2();
            else                     wait_async_le0();
        }

        const float zr = __shfl((i < 32) ? zA : zB, i & 31, 32);
        const float4 g0 = *(const float4*)(&gbuf[w][i & (PIPE - 1)][4 * lane]);
        const float4 g1 = *(const float4*)(&gbuf[w][i & (PIPE - 1)][128 + 4 * lane]);
        float g[8] = {g0.x, g0.y, g0.z, g0.w, g1.x, g1.y, g1.z, g1.w};

        float dist[8], t[8];
#pragma unroll
        for (int j = 0; j < 8; ++j) {
            dist[j] = fabsf(zr - cb[j]);
            t[j]    = (g[j] - dist[j]) * inv_tau;
        }

        // wave32 max-reduce for a numerically stable softmax
        float m = t[0];
#pragma unroll
        for (int j = 1; j < 8; ++j) m = fmaxf(m, t[j]);
#pragma unroll
        for (int s = 16; s >= 1; s >>= 1) m = fmaxf(m, __shfl_xor(m, s, 32));

        float e[8], ssum = 0.f, dot = 0.f;
#pragma unroll
        for (int j = 0; j < 8; ++j) {
            e[j] = __expf(t[j] - m);
            ssum += e[j];
            dot  += e[j] * cb[j];
        }
#pragma unroll
        for (int s = 16; s >= 1; s >>= 1) {
            ssum += __shfl_xor(ssum, s, 32);
            dot  += __shfl_xor(dot, s, 32);
        }
        const float inv_s = 1.0f / ssum;

#pragma unroll
        for (int j = 0; j < 8; ++j) acc[j] += e[j] * inv_s;

        // argmin over dist; first-occurrence tie-break (smaller k wins)
        float dmin = dist[0];
        int   kmin = kk[0];
#pragma unroll
        for (int j = 1; j < 8; ++j)
            if (dist[j] < dmin) { dmin = dist[j]; kmin = kk[j]; }
        unsigned long long key =
            ((unsigned long long)__float_as_uint(dmin) << 32) | (unsigned)kmin;
#pragma unroll
        for (int s = 16; s >= 1; s >>= 1) {
            unsigned long long o = __shfl_xor(key, s, 32);
            key = (o < key) ? o : key;
        }

        if (lane == 0) {
            out[row]         = dot * inv_s;
            out[N + 1 + row] = (float)(unsigned)(key & 0xFFFFFFFFull);
        }
    }

    // deterministic per-block merge of probability sums (reuse staging LDS;
    // every wave has drained its own async ops via wait_async_le0 above)
    __syncthreads();
#pragma unroll
    for (int j = 0; j < 8; ++j) accbuf[w][kk[j]] = acc[j];
    __syncthreads();
    float ps = 0.f;
#pragma unroll
    for (int ww = 0; ww < WAVES_PER_BLOCK; ++ww) ps += accbuf[ww][tid];
    partial[blockIdx.x * KCODES + tid] = ps;
}

// ---- finalize: avg_probs -> perplexity (single block, deterministic) -------
__global__ void __launch_bounds__(BLOCK_THREADS)
gumbel_vq_finalize(const float* __restrict__ partial, float* __restrict__ out,
                   int nblk, int N) {
    const int t = threadIdx.x;
    float s = 0.f;
    for (int b = 0; b < nblk; ++b) s += partial[b * KCODES + t];
    const float avg = s / (float)N;
    __shared__ float red[KCODES];
    red[t] = avg * logf(avg + 1e-10f);
    __syncthreads();
    for (int stride = KCODES / 2; stride > 0; stride >>= 1) {
        if (t < stride) red[t] += red[t + stride];
        __syncthreads();
    }
    if (t == 0) out[N] = expf(-red[0]);
}

// ---- launch ----------------------------------------------------------------
extern "C" void kernel_launch(void* const* d_in, const int* in_sizes, int n_in,
                              void* d_out, int out_size, void* d_ws, size_t ws_size,
                              hipStream_t stream) {
    const float* z        = (const float*)d_in[0];
    const float* codebook = (const float*)d_in[1];
    const float* log_temp = (const float*)d_in[2];
    const float* gumbel   = (const float*)d_in[3];
    float* out            = (float*)d_out;
    float* partial        = (float*)d_ws;

    const int N    = in_sizes[0];             // 262144
    const int nblk = N / ROWS_PER_BLOCK;      // 512 blocks -> 512KB of d_ws

    gumbel_vq_main<<<nblk, BLOCK_THREADS, 0, stream>>>(
        z, codebook, log_temp, gumbel, out, partial, N);
    gumbel_vq_finalize<<<1, BLOCK_THREADS, 0, stream>>>(partial, out, nblk, N);
}